// RbfNet_75033078661354
// MI455X (gfx1250) — compile-verified
//
#include <hip/hip_runtime.h>
#include <hip/hip_bf16.h>

typedef float v2f __attribute__((ext_vector_type(2)));
typedef float v8f __attribute__((ext_vector_type(8)));

#define N_PART 50000
#define N_EDGE 800000

// ---------------------------------------------------------------------------
// Kernel 1: per-edge tent-basis factorization.
// 8 tent centers on linspace(-1,1): for any clamped x only slots k0,k0+1 are
// nonzero with weights (1-frac, frac). Self-edges get zero weight.
// ---------------------------------------------------------------------------
__global__ __launch_bounds__(256) void rbf_basis_kernel(
    const float* __restrict__ dist, const int* __restrict__ fi,
    const int* __restrict__ fj, int* __restrict__ kbuf,
    float* __restrict__ wbuf, int E) {
  int e = blockIdx.x * blockDim.x + threadIdx.x;
  if (e >= E) return;
  float x = fminf(1.0f, fmaxf(-1.0f, dist[e]));
  float t = (x + 1.0f) * 3.5f;  // 1/h with h = 2/7
  int k0 = (int)t;
  k0 = k0 < 0 ? 0 : (k0 > 6 ? 6 : k0);
  float frac = t - (float)k0;
  float m = (fi[e] != fj[e]) ? 1.0f : 0.0f;
  kbuf[e] = k0;
  wbuf[2 * e + 0] = (1.0f - frac) * m;
  wbuf[2 * e + 1] = frac * m;
}

// ---------------------------------------------------------------------------
// Kernel 2: weight pre-pack for the WMMA B fragment.
// B[k][col] -> Bp[((k/2)*ncols + col)*2 + (k&1)], so a lane's fragment
// (rows 2h,2h+1 at one column) is a single contiguous b64 load.
// ---------------------------------------------------------------------------
__global__ __launch_bounds__(256) void pack_pairs_kernel(
    const float* __restrict__ B, float* __restrict__ Bp, int K, int ncols) {
  int i = blockIdx.x * blockDim.x + threadIdx.x;
  if (i >= K * ncols) return;
  int k = i / ncols, col = i - k * ncols;
  Bp[(((k >> 1) * ncols + col) << 1) + (k & 1)] = B[i];
}

// ---------------------------------------------------------------------------
// Kernel 3: edge scatter (dominant cost). One wave32 per edge. Per-edge
// scalars forced wave-uniform (readfirstlane) -> scalar loads + scalar
// early-outs. CIN templated: 64 -> b64 gather + 4 f32 global atomics/lane.
// agg layout [N, 8, CIN] == virtual A matrix [N, 8*CIN].
// ---------------------------------------------------------------------------
template <int CIN, int RELU>
__global__ __launch_bounds__(256) void rbf_scatter_kernel(
    const float* __restrict__ src, const int* __restrict__ fi,
    const int* __restrict__ fj, const int* __restrict__ kbuf,
    const float* __restrict__ wbuf, float* __restrict__ agg, int E) {
  int g = blockIdx.x * blockDim.x + threadIdx.x;
  int e = __builtin_amdgcn_readfirstlane(g >> 5);  // wave-uniform edge id
  if (e >= E) return;
  int lane = threadIdx.x & 31;
  float w0 = wbuf[2 * e + 0];
  float w1 = wbuf[2 * e + 1];
  if (w0 == 0.0f && w1 == 0.0f) return;  // self-edge
  int k0 = kbuf[e];
  const float* s = src + (size_t)fj[e] * CIN;
  float* d0 = agg + ((size_t)fi[e] * 8 + k0) * CIN;
  float* d1 = d0 + CIN;  // slot k0+1 (k0 <= 6 always)
  if (CIN == 64) {
    int c = lane * 2;
    v2f v = *(const v2f*)(s + c);
    if (RELU) { v.x = fmaxf(v.x, 0.0f); v.y = fmaxf(v.y, 0.0f); }
    if (w0 != 0.0f) {
      unsafeAtomicAdd(d0 + c + 0, w0 * v.x);
      unsafeAtomicAdd(d0 + c + 1, w0 * v.y);
    }
    if (w1 != 0.0f) {
      unsafeAtomicAdd(d1 + c + 0, w1 * v.x);
      unsafeAtomicAdd(d1 + c + 1, w1 * v.y);
    }
  } else {
    int c = lane;
    float v = s[c];
    if (RELU) v = fmaxf(v, 0.0f);
    if (w0 != 0.0f) unsafeAtomicAdd(d0 + c, w0 * v);
    if (w1 != 0.0f) unsafeAtomicAdd(d1 + c, w1 * v);
  }
}

// ---------------------------------------------------------------------------
// Kernel 4: fused FP32-WMMA GEMM per layer.
//   out[:, col_off:+NCOLS] = A0 @ B0 + act(A1) @ B1 + bias0 + bias1 (+resid)
// One wave -> 16 x (16*NT) tile via V_WMMA_F32_16X16X4_F32.
// A frag/lane: row = lane&15, k = kbase + 2*(lane>>4) + {0,1} (b64 load).
// B frag/lane: packed pair (b64 load). OOB handled by clamp+mask multiplies,
// never by EXEC divergence, so EXEC is all-ones through every WMMA.
// ---------------------------------------------------------------------------
template <int NT, int NCOLS, int RELU1>
__global__ __launch_bounds__(128) void rbf_gemm_kernel(
    const float* __restrict__ A0, int K0, const float* __restrict__ B0p,
    const float* __restrict__ A1, int K1, const float* __restrict__ B1p,
    const float* __restrict__ bias0, const float* __restrict__ bias1,
    const float* __restrict__ resid, int ldres, float* __restrict__ out,
    int ldout, int col_off, int nrows) {
  const int lane = threadIdx.x & 31;
  const int wave = threadIdx.x >> 5;
  const int m = lane & 15;
  const int half = lane >> 4;
  const int r0 = (blockIdx.x * 4 + wave) * 16;

  int colc[NT];
  float cmask[NT];
#pragma unroll
  for (int t = 0; t < NT; ++t) {  // compile-time trivial when 16*NT == NCOLS
    int col = t * 16 + m;
    bool ok = (col < NCOLS);
    colc[t] = ok ? col : (NCOLS - 1);
    cmask[t] = ok ? 1.0f : 0.0f;
  }

  v8f acc[NT] = {};
  const int arow = (r0 + m < nrows) ? (r0 + m) : 0;
  const float rowmask = (r0 + m < nrows) ? 1.0f : 0.0f;

  if (A0) {  // conv path: agg[N, K0] @ conv_w[K0, NCOLS]
    const float* ap = A0 + (size_t)arow * K0;
    for (int k = 0; k < K0; k += 4) {
      v2f a = *(const v2f*)(ap + k + 2 * half);
      a.x *= rowmask;
      a.y *= rowmask;
      const float* bp = B0p + ((size_t)((k >> 1) + half) * NCOLS) * 2;
#pragma unroll
      for (int t = 0; t < NT; ++t) {
        v2f b = *(const v2f*)(bp + colc[t] * 2);
        b.x *= cmask[t];
        b.y *= cmask[t];
        acc[t] = __builtin_amdgcn_wmma_f32_16x16x4_f32(
            false, a, false, b, (short)0, acc[t], false, false);
      }
    }
  }
  if (A1) {  // dense path: act(prev)[N, K1] @ fc_w[K1, NCOLS]
    const float* ap = A1 + (size_t)arow * K1;
    for (int k = 0; k < K1; k += 4) {
      v2f a = *(const v2f*)(ap + k + 2 * half);
      if (RELU1) {
        a.x = fmaxf(a.x, 0.0f);
        a.y = fmaxf(a.y, 0.0f);
      }
      a.x *= rowmask;
      a.y *= rowmask;
      const float* bp = B1p + ((size_t)((k >> 1) + half) * NCOLS) * 2;
#pragma unroll
      for (int t = 0; t < NT; ++t) {
        v2f b = *(const v2f*)(bp + colc[t] * 2);
        b.x *= cmask[t];
        b.y *= cmask[t];
        acc[t] = __builtin_amdgcn_wmma_f32_16x16x4_f32(
            false, a, false, b, (short)0, acc[t], false, false);
      }
    }
  }

  // epilogue: biases, residual, store (divergence allowed; WMMAs are done)
#pragma unroll
  for (int t = 0; t < NT; ++t) {
    int col = t * 16 + m;
    if (col >= NCOLS) continue;
    float bsum = 0.0f;
    if (bias0) bsum += bias0[col];
    if (bias1) bsum += bias1[col];
#pragma unroll
    for (int r = 0; r < 8; ++r) {
      int row = r0 + r + 8 * half;
      if (row < nrows) {
        float v = acc[t][r] + bsum;
        if (resid) v += resid[(size_t)row * ldres + col];
        out[(size_t)row * ldout + col_off + col] = v;
      }
    }
  }
}

// ---------------------------------------------------------------------------
// Host-side pipeline (all on `stream`, graph-capture safe).
// ---------------------------------------------------------------------------
extern "C" void kernel_launch(void* const* d_in, const int* in_sizes, int n_in,
                              void* d_out, int out_size, void* d_ws,
                              size_t ws_size, hipStream_t stream) {
  const float* x = (const float*)d_in[0];     // [N,32]
  const int* fi = (const int*)d_in[1];        // [E]
  const int* fj = (const int*)d_in[2];        // [E]
  const float* dist = (const float*)d_in[3];  // [E,1]
  const float *cw[4], *cb[4], *fw[4], *fb[4];
  for (int i = 0; i < 4; ++i) {
    cw[i] = (const float*)d_in[4 + i];
    cb[i] = (const float*)d_in[8 + i];
    fw[i] = (const float*)d_in[12 + i];
    fb[i] = (const float*)d_in[16 + i];
  }
  float* outp = (float*)d_out;

  // Workspace carve-up (256B aligned offsets).
  char* ws = (char*)d_ws;
  float* agg = (float*)ws;                            // N*8*64 f32 = 102.4MB
  float* bufA = (float*)(ws + 102400000);             // N*64 f32
  float* bufB = (float*)(ws + 115200000);             // N*64 f32
  int* kbuf = (int*)(ws + 128000000);                 // E i32
  float* wbuf = (float*)(ws + 131200000);             // E*2 f32
  // packed weights (k-pair layout)
  float* cwp[4], *fwp[4];
  size_t off = 137600000;
  const int cK[4] = {256, 512, 512, 512};             // 8*cin
  const int dK[4] = {32, 64, 64, 64};                 // cin
  const int co[4] = {32, 64, 64, 2};                  // cout
  for (int i = 0; i < 4; ++i) {
    cwp[i] = (float*)(ws + off);
    off += (size_t)cK[i] * co[i] * 4;
    off = (off + 255) & ~(size_t)255;
  }
  for (int i = 0; i < 4; ++i) {
    fwp[i] = (float*)(ws + off);
    off += (size_t)dK[i] * co[i] * 4;
    off = (off + 255) & ~(size_t)255;
  }

  const int E = N_EDGE, N = N_PART;
  const int gemmBlocks = (N + 63) / 64;
  const int scatBlocks = (E * 32) / 256;

  // Per-edge basis factorization (reused by all 4 layers).
  rbf_basis_kernel<<<(E + 255) / 256, 256, 0, stream>>>(dist, fi, fj, kbuf,
                                                        wbuf, E);
  // Pack all weight matrices into WMMA-friendly pair layout.
  for (int i = 0; i < 4; ++i) {
    int n0 = cK[i] * co[i];
    pack_pairs_kernel<<<(n0 + 255) / 256, 256, 0, stream>>>(cw[i], cwp[i],
                                                            cK[i], co[i]);
    int n1 = dK[i] * co[i];
    pack_pairs_kernel<<<(n1 + 255) / 256, 256, 0, stream>>>(fw[i], fwp[i],
                                                            dK[i], co[i]);
  }

  // ---- layer 0: hstack(fc(x), conv(x)) -> bufA [N,64] ----
  hipMemsetAsync(agg, 0, (size_t)N * 8 * 32 * sizeof(float), stream);
  rbf_scatter_kernel<32, 0><<<scatBlocks, 256, 0, stream>>>(x, fi, fj, kbuf,
                                                            wbuf, agg, E);
  rbf_gemm_kernel<2, 32, 0><<<gemmBlocks, 128, 0, stream>>>(
      nullptr, 0, nullptr, x, 32, fwp[0], nullptr, fb[0], nullptr, 0, bufA, 64,
      0, N);
  rbf_gemm_kernel<2, 32, 0><<<gemmBlocks, 128, 0, stream>>>(
      agg, 256, cwp[0], nullptr, 0, nullptr, cb[0], nullptr, nullptr, 0, bufA,
      64, 32, N);

  // ---- layer 1: bufA -> bufB [N,64] ----
  hipMemsetAsync(agg, 0, (size_t)N * 8 * 64 * sizeof(float), stream);
  rbf_scatter_kernel<64, 1><<<scatBlocks, 256, 0, stream>>>(bufA, fi, fj, kbuf,
                                                            wbuf, agg, E);
  rbf_gemm_kernel<4, 64, 1><<<gemmBlocks, 128, 0, stream>>>(
      agg, 512, cwp[1], bufA, 64, fwp[1], cb[1], fb[1], nullptr, 0, bufB, 64, 0,
      N);

  // ---- layer 2 (residual): bufB -> bufA [N,64] ----
  hipMemsetAsync(agg, 0, (size_t)N * 8 * 64 * sizeof(float), stream);
  rbf_scatter_kernel<64, 1><<<scatBlocks, 256, 0, stream>>>(bufB, fi, fj, kbuf,
                                                            wbuf, agg, E);
  rbf_gemm_kernel<4, 64, 1><<<gemmBlocks, 128, 0, stream>>>(
      agg, 512, cwp[2], bufB, 64, fwp[2], cb[2], fb[2], bufB, 64, bufA, 64, 0,
      N);

  // ---- layer 3: bufA -> d_out [N,2] ----
  hipMemsetAsync(agg, 0, (size_t)N * 8 * 64 * sizeof(float), stream);
  rbf_scatter_kernel<64, 1><<<scatBlocks, 256, 0, stream>>>(bufA, fi, fj, kbuf,
                                                            wbuf, agg, E);
  rbf_gemm_kernel<1, 2, 1><<<gemmBlocks, 128, 0, stream>>>(
      agg, 512, cwp[3], bufA, 64, fwp[3], cb[3], fb[3], nullptr, 0, outp, 2, 0,
      N);

  (void)in_sizes; (void)n_in; (void)out_size; (void)ws_size;
}